// DrugGAT_GCN_26671746908431
// MI455X (gfx1250) — compile-verified
//
#include <hip/hip_runtime.h>
#include <hip/hip_bf16.h>
#include <math.h>

// ---------------- problem constants (match reference) ----------------
#define NN   30000
#define EE   240000
#define GG   1000
#define FF   78
#define HH   10
#define HID  780      // F*H
#define FC1  1500
#define FC2  128
#define POOL2 1560    // 2*HID
#define NEG_SLOPE 0.2f

typedef __attribute__((ext_vector_type(2))) float v2f;
typedef __attribute__((ext_vector_type(8))) float v8f;

#if defined(__AMDGCN__)
#define WMMA4(acc_, a_, b_) \
    (acc_) = __builtin_amdgcn_wmma_f32_16x16x4_f32( \
        false, (a_), false, (b_), (short)0, (acc_), false, false)
#else
#define WMMA4(acc_, a_, b_) ((void)0)
#endif

// ---------------- helpers ----------------
__device__ __forceinline__ void atomicMaxF32(float* addr, float val) {
    // IEEE-754 ordering trick: works for mixed-sign values incl. -inf init.
    if (val >= 0.0f) {
        atomicMax((int*)addr, __float_as_int(val));
    } else {
        atomicMin((unsigned int*)addr, __float_as_uint(val));
    }
}

__device__ __forceinline__ float lrelu(float v) {
    return v > 0.0f ? v : NEG_SLOPE * v;
}

// ---------------- guarded 16x16 tile (edge path) ----------------
__device__ __forceinline__ void gemm_tile_guarded(
    const float* __restrict__ A, const float* __restrict__ B,
    const float* __restrict__ bias, float* __restrict__ C,
    int M, int N, int K, int relu, int tm, int tn, int lane)
{
    if (tm >= M || tn >= N) return;          // wave-uniform
    const int half = lane >> 4;
    const int l    = lane & 15;
    const int ra = tm + l;
    const int cb = tn + l;
    const bool rowOK = (ra < M);
    const bool colOK = (cb < N);
    v8f acc = {0.f,0.f,0.f,0.f,0.f,0.f,0.f,0.f};
    for (int k0 = 0; k0 < K; k0 += 4) {
        const int ka = k0 + (half << 1);
        v2f a, b;
        a.x = (rowOK && ka     < K) ? A[(size_t)ra * K + ka]       : 0.f;
        a.y = (rowOK && ka + 1 < K) ? A[(size_t)ra * K + ka + 1]   : 0.f;
        b.x = (colOK && ka     < K) ? B[(size_t)ka * N + cb]       : 0.f;
        b.y = (colOK && ka + 1 < K) ? B[(size_t)(ka + 1) * N + cb] : 0.f;
        WMMA4(acc, a, b);
    }
    if (colOK) {
        const float bv = bias ? bias[cb] : 0.f;
#pragma unroll
        for (int v = 0; v < 8; ++v) {
            const int row = tm + (half << 3) + v;
            if (row < M) {
                float val = acc[v] + bv;
                if (relu) val = fmaxf(val, 0.f);
                C[(size_t)row * N + cb] = val;
            }
        }
    }
}

// ---------------- fp32 WMMA GEMM: C = act(A[M,K] @ B[K,N] + bias) ----------------
// Row-major. One wave32 computes a 64(M) x 32(N) strip: 4x2 register tiles of
// 16x16 via V_WMMA_F32_16X16X4_F32. Interior strips take a guard-free fast path
// (vector b64 A loads, global_prefetch on the K stream); boundary strips take
// the guarded scalar path. Requires K even (all call sites satisfy).
__global__ void wmma_gemm_f32(const float* __restrict__ A,
                              const float* __restrict__ B,
                              const float* __restrict__ bias,
                              float* __restrict__ C,
                              int M, int N, int K, int relu)
{
    const int lane = threadIdx.x & 31;
    const int half = lane >> 4;
    const int l    = lane & 15;
    const int wavesPerBlock = blockDim.x >> 5;
    const int waveId  = blockIdx.x * wavesPerBlock + (threadIdx.x >> 5);
    const int wTilesN = (N + 31) >> 5;
    const int wTilesM = (M + 63) >> 6;
    if (waveId >= wTilesM * wTilesN) return;     // uniform per wave
    const int wm = (waveId / wTilesN) << 6;
    const int wn = (waveId % wTilesN) << 5;

    if (wm + 64 <= M && wn + 32 <= N) {
        // -------- fast interior path --------
        v8f acc00 = {0.f,0.f,0.f,0.f,0.f,0.f,0.f,0.f};
        v8f acc01 = acc00, acc10 = acc00, acc11 = acc00;
        v8f acc20 = acc00, acc21 = acc00, acc30 = acc00, acc31 = acc00;

        const size_t Ks = (size_t)K;
        const float* A0 = A + (size_t)(wm + l) * Ks;           // rows wm+l+16t
        const int cb0 = wn + l;
        const int cb1 = wn + 16 + l;
        const int K4 = K & ~3;

        for (int k0 = 0; k0 < K4; k0 += 4) {
            const int ka = k0 + (half << 1);
            const v2f a0 = *(const v2f*)(A0 +  0 * Ks + ka);
            const v2f a1 = *(const v2f*)(A0 + 16 * Ks + ka);
            const v2f a2 = *(const v2f*)(A0 + 32 * Ks + ka);
            const v2f a3 = *(const v2f*)(A0 + 48 * Ks + ka);
            const float* Br0 = B + (size_t)ka * N;
            const float* Br1 = Br0 + N;
            v2f b0, b1;
            b0.x = Br0[cb0]; b0.y = Br1[cb0];
            b1.x = Br0[cb1]; b1.y = Br1[cb1];
            // prefetch the K-stream ahead (L2-resident weights stream into WGP$)
            __builtin_prefetch(A0 + ka + 32, 0, 1);
            __builtin_prefetch(Br0 + 4 * N + cb0, 0, 1);
            WMMA4(acc00, a0, b0);
            WMMA4(acc01, a0, b1);
            WMMA4(acc10, a1, b0);
            WMMA4(acc11, a1, b1);
            WMMA4(acc20, a2, b0);
            WMMA4(acc21, a2, b1);
            WMMA4(acc30, a3, b0);
            WMMA4(acc31, a3, b1);
        }
        if (K4 < K) {                           // K remainder (e.g. K=78 -> 2)
            const int ka = K4 + (half << 1);
            v2f a0, a1, a2, a3, b0, b1;
            a0.x = (ka < K) ? A0[ 0 * Ks + ka] : 0.f;
            a0.y = (ka + 1 < K) ? A0[ 0 * Ks + ka + 1] : 0.f;
            a1.x = (ka < K) ? A0[16 * Ks + ka] : 0.f;
            a1.y = (ka + 1 < K) ? A0[16 * Ks + ka + 1] : 0.f;
            a2.x = (ka < K) ? A0[32 * Ks + ka] : 0.f;
            a2.y = (ka + 1 < K) ? A0[32 * Ks + ka + 1] : 0.f;
            a3.x = (ka < K) ? A0[48 * Ks + ka] : 0.f;
            a3.y = (ka + 1 < K) ? A0[48 * Ks + ka + 1] : 0.f;
            b0.x = (ka < K) ? B[(size_t)ka * N + cb0] : 0.f;
            b0.y = (ka + 1 < K) ? B[(size_t)(ka + 1) * N + cb0] : 0.f;
            b1.x = (ka < K) ? B[(size_t)ka * N + cb1] : 0.f;
            b1.y = (ka + 1 < K) ? B[(size_t)(ka + 1) * N + cb1] : 0.f;
            WMMA4(acc00, a0, b0);
            WMMA4(acc01, a0, b1);
            WMMA4(acc10, a1, b0);
            WMMA4(acc11, a1, b1);
            WMMA4(acc20, a2, b0);
            WMMA4(acc21, a2, b1);
            WMMA4(acc30, a3, b0);
            WMMA4(acc31, a3, b1);
        }

        const float bv0 = bias ? bias[cb0] : 0.f;
        const float bv1 = bias ? bias[cb1] : 0.f;
        v8f* accs[4][2] = {{&acc00,&acc01},{&acc10,&acc11},{&acc20,&acc21},{&acc30,&acc31}};
#pragma unroll
        for (int t = 0; t < 4; ++t) {
#pragma unroll
            for (int v = 0; v < 8; ++v) {
                const int row = wm + (t << 4) + (half << 3) + v;
                float f0 = (*accs[t][0])[v] + bv0;
                float f1 = (*accs[t][1])[v] + bv1;
                if (relu) { f0 = fmaxf(f0, 0.f); f1 = fmaxf(f1, 0.f); }
                float* crow = C + (size_t)row * N;
                crow[cb0] = f0;
                crow[cb1] = f1;
            }
        }
    } else {
        // -------- guarded edge path --------
#pragma unroll
        for (int t = 0; t < 4; ++t)
#pragma unroll
            for (int u = 0; u < 2; ++u)
                gemm_tile_guarded(A, B, bias, C, M, N, K, relu,
                                  wm + (t << 4), wn + (u << 4), lane);
    }
}

// ---------------- fill ----------------
__global__ void fill_f32(float* __restrict__ p, float v, size_t n)
{
    size_t i = (size_t)blockIdx.x * blockDim.x + threadIdx.x;
    if (i < n) p[i] = v;
}

// ---------------- per-(node,head) attention logits ----------------
__global__ void attn_proj(const float* __restrict__ h,
                          const float* __restrict__ att_src,
                          const float* __restrict__ att_dst,
                          float* __restrict__ aS, float* __restrict__ aD)
{
    int i = blockIdx.x * blockDim.x + threadIdx.x;   // over N*H
    if (i >= NN * HH) return;
    int n = i / HH, hh = i % HH;
    const float* hp = h + (size_t)n * HID + hh * FF;
    const float* as = att_src + hh * FF;
    const float* ad = att_dst + hh * FF;
    float s1 = 0.f, s2 = 0.f;
#pragma unroll 6
    for (int f = 0; f < FF; ++f) {
        float v = hp[f];
        s1 += v * as[f];
        s2 += v * ad[f];
    }
    aS[i] = s1;
    aD[i] = s2;
}

// ---------------- edge pass 1: segment max of leaky-relu logits ----------------
__global__ void edge_attn_max(const int* __restrict__ src, const int* __restrict__ dst,
                              const float* __restrict__ aS, const float* __restrict__ aD,
                              float* __restrict__ emax)
{
    int e = blockIdx.x * blockDim.x + threadIdx.x;
    const int Et = EE + NN;
    if (e >= Et) return;
    int s = (e < EE) ? src[e] : (e - EE);
    int d = (e < EE) ? dst[e] : (e - EE);
#pragma unroll
    for (int hh = 0; hh < HH; ++hh) {
        float v = lrelu(aS[s * HH + hh] + aD[d * HH + hh]);
        atomicMaxF32(&emax[d * HH + hh], v);
    }
}

// ---------------- edge pass 2: segment sum of exp + in-degree ----------------
__global__ void edge_attn_sum(const int* __restrict__ src, const int* __restrict__ dst,
                              const float* __restrict__ aS, const float* __restrict__ aD,
                              const float* __restrict__ emax,
                              float* __restrict__ denom, float* __restrict__ deg)
{
    int e = blockIdx.x * blockDim.x + threadIdx.x;
    const int Et = EE + NN;
    if (e >= Et) return;
    int s = (e < EE) ? src[e] : (e - EE);
    int d = (e < EE) ? dst[e] : (e - EE);
#pragma unroll
    for (int hh = 0; hh < HH; ++hh) {
        float v = lrelu(aS[s * HH + hh] + aD[d * HH + hh]);
        atomicAdd(&denom[d * HH + hh], __expf(v - emax[d * HH + hh]));
    }
    atomicAdd(&deg[d], 1.0f);
}

// ---------------- edge pass 3: alpha-weighted message scatter ----------------
__global__ void edge_gat_scatter(const int* __restrict__ src, const int* __restrict__ dst,
                                 const float* __restrict__ aS, const float* __restrict__ aD,
                                 const float* __restrict__ emax, const float* __restrict__ denom,
                                 const float* __restrict__ h, float* __restrict__ out)
{
    int e = blockIdx.x;
    int s = (e < EE) ? src[e] : (e - EE);
    int d = (e < EE) ? dst[e] : (e - EE);
    __shared__ float alpha[HH];
    if (threadIdx.x < HH) {
        int hh = threadIdx.x;
        float v = lrelu(aS[s * HH + hh] + aD[d * HH + hh]);
        alpha[hh] = __expf(v - emax[d * HH + hh]) / denom[d * HH + hh];
    }
    __syncthreads();
    const float* hp = h + (size_t)s * HID;
    float* op = out + (size_t)d * HID;
    for (int f = threadIdx.x; f < HID; f += blockDim.x) {
        atomicAdd(&op[f], hp[f] * alpha[f / FF]);
    }
}

// ---------------- bias + relu (elementwise, broadcast bias over rows) ----------------
__global__ void bias_relu(float* __restrict__ x, const float* __restrict__ b,
                          int rows, int cols)
{
    size_t i = (size_t)blockIdx.x * blockDim.x + threadIdx.x;
    if (i >= (size_t)rows * cols) return;
    int c = (int)(i % cols);
    x[i] = fmaxf(x[i] + b[c], 0.f);
}

// ---------------- GCN: norm-weighted scatter ----------------
__global__ void edge_gcn_scatter(const int* __restrict__ src, const int* __restrict__ dst,
                                 const float* __restrict__ deg,
                                 const float* __restrict__ h2, float* __restrict__ out2)
{
    int e = blockIdx.x;
    int s = (e < EE) ? src[e] : (e - EE);
    int d = (e < EE) ? dst[e] : (e - EE);
    float norm = rsqrtf(fmaxf(deg[s], 1.f)) * rsqrtf(fmaxf(deg[d], 1.f));
    const float* hp = h2 + (size_t)s * HID;
    float* op = out2 + (size_t)d * HID;
    for (int f = threadIdx.x; f < HID; f += blockDim.x) {
        atomicAdd(&op[f], norm * hp[f]);
    }
}

// ---------------- pooling: per-graph sum / max / count ----------------
__global__ void pool_scatter(const float* __restrict__ out2, const int* __restrict__ batch,
                             float* __restrict__ gsum, float* __restrict__ gmax,
                             float* __restrict__ counts)
{
    int n = blockIdx.x;
    int g = batch[n];
    if (threadIdx.x == 0) atomicAdd(&counts[g], 1.0f);
    const float* xp = out2 + (size_t)n * HID;
    for (int f = threadIdx.x; f < HID; f += blockDim.x) {
        float v = xp[f];
        atomicAdd(&gsum[(size_t)g * HID + f], v);
        atomicMaxF32(&gmax[(size_t)g * HID + f], v);
    }
}

__global__ void pool_finalize(const float* __restrict__ gsum, const float* __restrict__ gmax,
                              const float* __restrict__ counts, float* __restrict__ pooled)
{
    int i = blockIdx.x * blockDim.x + threadIdx.x;  // over G*HID
    if (i >= GG * HID) return;
    int g = i / HID, f = i % HID;
    float c = counts[g];
    float mean = gsum[i] / fmaxf(c, 1.f);
    float mx = (c > 0.f) ? gmax[i] : 0.f;
    pooled[(size_t)g * POOL2 + f] = mean;
    pooled[(size_t)g * POOL2 + HID + f] = mx;
}

// ---------------- host side ----------------
static inline void launch_gemm(const float* A, const float* B, const float* bias, float* C,
                               int M, int N, int K, int relu, hipStream_t stream)
{
    int waveTiles = ((M + 63) / 64) * ((N + 31) / 32);
    int blocks = (waveTiles + 7) / 8;   // 8 waves of 32 per 256-thread block
    hipLaunchKernelGGL(wmma_gemm_f32, dim3(blocks), dim3(256), 0, stream,
                       A, B, bias, C, M, N, K, relu);
}

static inline void launch_fill(float* p, float v, size_t n, hipStream_t stream)
{
    size_t blocks = (n + 255) / 256;
    hipLaunchKernelGGL(fill_f32, dim3((unsigned)blocks), dim3(256), 0, stream, p, v, n);
}

extern "C" void kernel_launch(void* const* d_in, const int* in_sizes, int n_in,
                              void* d_out, int out_size, void* d_ws, size_t ws_size,
                              hipStream_t stream)
{
    const float* x       = (const float*)d_in[0];
    const int*   ei      = (const int*)  d_in[1];
    const int*   batch   = (const int*)  d_in[2];
    const float* W_gat   = (const float*)d_in[3];
    const float* att_src = (const float*)d_in[4];
    const float* att_dst = (const float*)d_in[5];
    const float* b_gat   = (const float*)d_in[6];
    const float* W_gcn   = (const float*)d_in[7];
    const float* b_gcn   = (const float*)d_in[8];
    const float* W1      = (const float*)d_in[9];
    const float* b1      = (const float*)d_in[10];
    const float* W2      = (const float*)d_in[11];
    const float* b2      = (const float*)d_in[12];
    float* outp = (float*)d_out;

    const int* src = ei;         // edge_index[0]
    const int* dst = ei + EE;    // edge_index[1]
    const int Et = EE + NN;

    // ---- workspace carve-up (floats) ----
    float* ws = (float*)d_ws;
    const size_t NHf = (size_t)NN * HID;
    float* bufA   = ws;                         // h, then h2
    float* bufB   = bufA + NHf;                 // GAT out, then GCN out2
    float* aS     = bufB + NHf;                 // [N,H]
    float* aD     = aS + (size_t)NN * HH;
    float* emax   = aD + (size_t)NN * HH;
    float* denom  = emax + (size_t)NN * HH;
    float* deg    = denom + (size_t)NN * HH;    // [N]
    float* counts = deg + NN;                   // [G]
    float* gsum   = counts + GG;                // [G,HID]
    float* gmax   = gsum + (size_t)GG * HID;    // [G,HID]
    float* pooled = gmax + (size_t)GG * HID;    // [G,2*HID]
    float* hidfc  = pooled + (size_t)GG * POOL2;// [G,FC1]

    // 1) h = x @ W_gat                                  [N,HID]
    launch_gemm(x, W_gat, nullptr, bufA, NN, HID, FF, 0, stream);

    // 2) per-(node,head) attention logits
    hipLaunchKernelGGL(attn_proj, dim3((NN * HH + 255) / 256), dim3(256), 0, stream,
                       bufA, att_src, att_dst, aS, aD);

    // 3) init accumulators
    launch_fill(emax, -INFINITY, (size_t)NN * HH, stream);
    launch_fill(denom, 0.f, (size_t)NN * HH, stream);
    launch_fill(deg, 0.f, NN, stream);
    launch_fill(bufB, 0.f, NHf, stream);

    // 4-6) edge passes: segment softmax + message scatter
    hipLaunchKernelGGL(edge_attn_max, dim3((Et + 255) / 256), dim3(256), 0, stream,
                       src, dst, aS, aD, emax);
    hipLaunchKernelGGL(edge_attn_sum, dim3((Et + 255) / 256), dim3(256), 0, stream,
                       src, dst, aS, aD, emax, denom, deg);
    hipLaunchKernelGGL(edge_gat_scatter, dim3(Et), dim3(256), 0, stream,
                       src, dst, aS, aD, emax, denom, bufA, bufB);

    // 7) out = relu(out + b_gat)
    hipLaunchKernelGGL(bias_relu, dim3((unsigned)((NHf + 255) / 256)), dim3(256), 0, stream,
                       bufB, b_gat, NN, HID);

    // 8) h2 = out @ W_gcn     (bufB -> bufA; bufA's h is dead now)
    launch_gemm(bufB, W_gcn, nullptr, bufA, NN, HID, HID, 0, stream);

    // 9) out2 = 0 (reuse bufB; GEMM above already consumed it, stream-ordered)
    launch_fill(bufB, 0.f, NHf, stream);

    // 10) out2 += norm * h2[src]
    hipLaunchKernelGGL(edge_gcn_scatter, dim3(Et), dim3(256), 0, stream,
                       src, dst, deg, bufA, bufB);

    // 11) out2 = relu(out2 + b_gcn)
    hipLaunchKernelGGL(bias_relu, dim3((unsigned)((NHf + 255) / 256)), dim3(256), 0, stream,
                       bufB, b_gcn, NN, HID);

    // 12) pooling accumulators
    launch_fill(counts, 0.f, GG, stream);
    launch_fill(gsum, 0.f, (size_t)GG * HID, stream);
    launch_fill(gmax, -INFINITY, (size_t)GG * HID, stream);

    // 13) per-node scatter into graph sum/max
    hipLaunchKernelGGL(pool_scatter, dim3(NN), dim3(256), 0, stream,
                       bufB, batch, gsum, gmax, counts);

    // 14) pooled = [mean | max]
    hipLaunchKernelGGL(pool_finalize, dim3((GG * HID + 255) / 256), dim3(256), 0, stream,
                       gsum, gmax, counts, pooled);

    // 15) hidfc = relu(pooled @ W1 + b1)
    launch_gemm(pooled, W1, b1, hidfc, GG, FC1, POOL2, 1, stream);

    // 16) d_out = hidfc @ W2 + b2
    launch_gemm(hidfc, W2, b2, outp, GG, FC2, FC1, 0, stream);
}